// GraphConvolution_53154515256135
// MI455X (gfx1250) — compile-verified
//
#include <hip/hip_runtime.h>

typedef __attribute__((ext_vector_type(2))) float v2f;
typedef __attribute__((ext_vector_type(8))) float v8f;

#define DFEAT 48   // D_IN == D_OUT == 48

// ---------------------------------------------------------------------------
// Kernel 1: xw = x @ w using V_WMMA_F32_16X16X4_F32.
// One wave32 computes a 16x48 output strip: 3 column tiles x 12 K-steps.
// A 16x4 f32 fragment: lanes 0-15 -> K={k,k+1}, lanes 16-31 -> K={k+2,k+3},
//                      M = lane&15, 2 VGPRs (v2f).
// B 4x16 f32 fragment: mirror with N = lane&15.
// C/D 16x16 f32: VGPR r holds (M = (lane>=16?8:0)+r, N = lane&15).
// ---------------------------------------------------------------------------
__global__ __launch_bounds__(256) void gemm_xw_wmma(
    const float* __restrict__ x, const float* __restrict__ w,
    float* __restrict__ xw, int n_nodes, int n_strips)
{
  const int wave = (int)((blockIdx.x * blockDim.x + threadIdx.x) >> 5);
  const int lane = (int)(threadIdx.x & 31);
  if (wave >= n_strips) return;        // wave-uniform: EXEC stays all-ones

  const int hi   = lane >> 4;          // half selector
  const int m    = lane & 15;
  const int row0 = wave * 16;

  // clamp load row so partial tail strips read in-bounds (stores are guarded)
  const int arow = (row0 + m < n_nodes) ? (row0 + m) : (n_nodes - 1);
  const float* xrow = x + (size_t)arow * DFEAT + hi * 2;

  v8f c0 = {}, c1 = {}, c2 = {};

  #pragma unroll
  for (int k = 0; k < DFEAT; k += 4) {
    // A fragment (8-byte aligned: row*48 + hi*2 + k is even)
    v2f a = *(const v2f*)(xrow + k);

    const int kb = k + hi * 2;
    v2f b0, b1, b2;
    b0.x = w[(kb + 0) * DFEAT + m];
    b0.y = w[(kb + 1) * DFEAT + m];
    b1.x = w[(kb + 0) * DFEAT + 16 + m];
    b1.y = w[(kb + 1) * DFEAT + 16 + m];
    b2.x = w[(kb + 0) * DFEAT + 32 + m];
    b2.y = w[(kb + 1) * DFEAT + 32 + m];

    // 8 args: (neg_a, A, neg_b, B, c_mod, C, reuse_a, reuse_b)
    c0 = __builtin_amdgcn_wmma_f32_16x16x4_f32(false, a, false, b0, (short)0, c0, false, false);
    c1 = __builtin_amdgcn_wmma_f32_16x16x4_f32(false, a, false, b1, (short)0, c1, false, false);
    c2 = __builtin_amdgcn_wmma_f32_16x16x4_f32(false, a, false, b2, (short)0, c2, false, false);
  }

  // Store D: lane writes rows row0 + hi*8 + r, column (tile*16 + m)
  #pragma unroll
  for (int r = 0; r < 8; ++r) {
    const int row = row0 + hi * 8 + r;
    if (row < n_nodes) {
      float* orow = xw + (size_t)row * DFEAT + m;
      orow[0]  = c0[r];
      orow[16] = c1[r];
      orow[32] = c2[r];
    }
  }
}

// ---------------------------------------------------------------------------
// Kernel 2: out[i] = bias[i % 48]  (bias folded into accumulator init)
// ---------------------------------------------------------------------------
__global__ __launch_bounds__(256) void init_bias_kernel(
    float* __restrict__ out, const float* __restrict__ b, int total)
{
  const int i = (int)(blockIdx.x * blockDim.x + threadIdx.x);
  if (i < total) out[i] = b[i % DFEAT];
}

// ---------------------------------------------------------------------------
// Kernel 3: COO scatter: out[dst] += val * xw[src]. 12 threads per edge,
// float4 gather per thread, 4 hardware fp32 atomics into L2-resident output.
// ---------------------------------------------------------------------------
__global__ __launch_bounds__(256) void spmm_scatter_kernel(
    const float* __restrict__ xw, const int* __restrict__ esrc,
    const int* __restrict__ edst, const float* __restrict__ evals,
    float* __restrict__ out, int n_edges)
{
  const long long tid = (long long)blockIdx.x * blockDim.x + threadIdx.x;
  const int e = (int)(tid / 12);
  if (e >= n_edges) return;
  const int f = (int)(tid % 12) * 4;

  const int   s = esrc[e];
  const int   d = edst[e];
  const float v = evals[e];

  const float4 msg = *(const float4*)(xw + (size_t)s * DFEAT + f);
  float* o = out + (size_t)d * DFEAT + f;
#if defined(__HIP_DEVICE_COMPILE__)
  unsafeAtomicAdd(o + 0, v * msg.x);
  unsafeAtomicAdd(o + 1, v * msg.y);
  unsafeAtomicAdd(o + 2, v * msg.z);
  unsafeAtomicAdd(o + 3, v * msg.w);
#else
  atomicAdd(o + 0, v * msg.x);
  atomicAdd(o + 1, v * msg.y);
  atomicAdd(o + 2, v * msg.z);
  atomicAdd(o + 3, v * msg.w);
#endif
}

// ---------------------------------------------------------------------------
// Kernel 4: in-place ReLU
// ---------------------------------------------------------------------------
__global__ __launch_bounds__(256) void relu_kernel(float* __restrict__ out, int total)
{
  const int i = (int)(blockIdx.x * blockDim.x + threadIdx.x);
  if (i < total) out[i] = fmaxf(out[i], 0.0f);
}

// ---------------------------------------------------------------------------
extern "C" void kernel_launch(void* const* d_in, const int* in_sizes, int n_in,
                              void* d_out, int out_size, void* d_ws, size_t ws_size,
                              hipStream_t stream)
{
  const float* x     = (const float*)d_in[0];  // [N, 48]
  const float* w     = (const float*)d_in[1];  // [48, 48]
  const float* b     = (const float*)d_in[2];  // [48]
  const int*   esrc  = (const int*)  d_in[3];  // [E]
  const int*   edst  = (const int*)  d_in[4];  // [E]
  const float* evals = (const float*)d_in[5];  // [E]

  const int n_nodes = in_sizes[0] / DFEAT;
  const int n_edges = in_sizes[3];

  float* out = (float*)d_out;          // [N, 48] fp32
  float* xw  = (float*)d_ws;           // scratch: N*48 floats

  // 1) dense GEMM via WMMA: one wave per 16-row strip
  const int n_strips = (n_nodes + 15) / 16;
  {
    const long long threads = (long long)n_strips * 32;
    const int blocks = (int)((threads + 255) / 256);
    gemm_xw_wmma<<<blocks, 256, 0, stream>>>(x, w, xw, n_nodes, n_strips);
  }

  // 2) init output accumulator with bias
  const int total = n_nodes * DFEAT;
  init_bias_kernel<<<(total + 255) / 256, 256, 0, stream>>>(out, b, total);

  // 3) edge scatter with fp32 atomics (12 threads / edge)
  {
    const long long threads = (long long)n_edges * 12;
    const int blocks = (int)((threads + 255) / 256);
    spmm_scatter_kernel<<<blocks, 256, 0, stream>>>(xw, esrc, edst, evals, out, n_edges);
  }

  // 4) ReLU in place
  relu_kernel<<<(total + 255) / 256, 256, 0, stream>>>(out, total);
}